// InstantNGP_81003083202712
// MI455X (gfx1250) — compile-verified
//
#include <hip/hip_runtime.h>
#include <stdint.h>

// ---------------- problem constants ----------------
#define BLOCK      512u
#define NPTS       (1u << 20)          // 1048576 points
#define NLVL       16
#define LDS_LEVELS 14                  // levels 0..13 staged in LDS
#define LDS_ROWS   ((1u << (LDS_LEVELS + 1)) - 2u)   // 32766 rows of float2 = 262128 B
#define ROWS_PER_WAVE 2048u            // 16 waves x 2048 rows covers 32766

static constexpr uint32_t P1 = 2654435761u;
static constexpr uint32_t P2 = 29675113u;
static constexpr uint32_t P3 = 123456789u;
// corner-delta constants: h(x+1,..) = h + P1*P2*P3 etc. (all mod 2^32)
static constexpr uint32_t C3 = P3;
static constexpr uint32_t C2 = (uint32_t)(P2 * P3);
static constexpr uint32_t C1 = (uint32_t)(P1 * C2);

typedef __attribute__((ext_vector_type(4))) unsigned int v4u_t;
typedef __attribute__((ext_vector_type(8))) int          v8i_t;
typedef __attribute__((ext_vector_type(4))) int          v4i_t;
typedef __attribute__((ext_vector_type(4))) float        v4f_t;

__global__ __launch_bounds__(BLOCK, 1)
void ngp_encode_kernel(const float* __restrict__ coords,
                       const float* __restrict__ table,
                       float* __restrict__ out)
{
    __shared__ float2 smem[LDS_ROWS];    // 262128 bytes of the 320 KB WGP LDS

    // -------- Stage hash-table levels 0..13 into LDS via the Tensor Data Mover.
    // 16 waves, each DMAs a 2048-row (16 KB) slice with its own 1-D descriptor.
    {
        uint32_t wave  = (uint32_t)__builtin_amdgcn_readfirstlane((int)(threadIdx.x >> 5));
        uint32_t start = wave * ROWS_PER_WAVE;               // row index
        uint32_t rem   = LDS_ROWS - start;
        uint32_t cnt   = rem < ROWS_PER_WAVE ? rem : ROWS_PER_WAVE;
        uint64_t gaddr = (uint64_t)(uintptr_t)table + (uint64_t)start * 8ull;
        uint32_t laddr = (uint32_t)(uintptr_t)(&smem[0]) + start * 8u;

#if __has_builtin(__builtin_amdgcn_tensor_load_to_lds)
        // D# group 0: count=1 | lds_addr | global_addr[56:0] | type=2
        v4u_t g0;
        g0.x = 1u;                                           // count = 1 (valid user desc)
        g0.y = laddr;                                        // LDS byte address
        g0.z = (uint32_t)gaddr;                              // global_addr[31:0]
        g0.w = (uint32_t)((gaddr >> 32) & 0x01FFFFFFull)     // global_addr[56:32]
             | 0x80000000u;                                  // type = 2 ("image")

        // D# group 1: data_size=8B(code 3), tensor_dim0 = tile_dim0 = cnt, 1-D tile
        v8i_t g1;
        g1[0] = (int)(3u << 16);                 // wg_mask=0, data_size=3 (8 bytes)
        g1[1] = (int)(cnt << 16);                // tensor_dim0[15:0] in bits[63:48]
        g1[2] = (int)(1u << 16);                 // tensor_dim0[31:16]=0, tensor_dim1=1
        g1[3] = (int)(cnt << 16);                // tile_dim0 in bits[127:112]
        g1[4] = 0;                               // tile_dim1 = tile_dim2 = 0 (unused)
        g1[5] = (int)cnt;                        // tensor_dim0_stride[31:0]
        g1[6] = 0;
        g1[7] = 0;

        v4i_t gz;  gz[0] = 0; gz[1] = 0; gz[2] = 0; gz[3] = 0;  // groups 2/3 unused
        v8i_t gz8; gz8[0] = 0; gz8[1] = 0; gz8[2] = 0; gz8[3] = 0;
                   gz8[4] = 0; gz8[5] = 0; gz8[6] = 0; gz8[7] = 0;

        // 6-arg form on this toolchain (clang-23 / therock-10.0):
        // (uint32x4 g0, int32x8 g1, int32x4 g2, int32x4 g3, int32x8, i32 cpol)
        __builtin_amdgcn_tensor_load_to_lds(g0, g1, gz, gz, gz8, 0);
        __builtin_amdgcn_s_wait_tensorcnt(0);
#else
        // Fallback: cooperative VALU staging
        for (uint32_t r = threadIdx.x; r < LDS_ROWS; r += BLOCK)
            smem[r] = ((const float2*)table)[r];
        (void)start; (void)cnt; (void)gaddr; (void)laddr;
#endif
    }
    __syncthreads();

    // -------- Per-point encoding --------
    const uint32_t p  = blockIdx.x * BLOCK + threadIdx.x;
    const float cx = coords[p * 3 + 0];
    const float cy = coords[p * 3 + 1];
    const float cz = coords[p * 3 + 2];

    constexpr uint32_t dC[8] = {0u, C1, C2, C1 + C2, C3, C1 + C3, C2 + C3, C1 + C2 + C3};

    float o[NLVL][2];

    auto setup = [&](float res, uint32_t& h000, float* w) {
        float sx = (cx + 1.0f) * res, sy = (cy + 1.0f) * res, sz = (cz + 1.0f) * res;
        float fx = floorf(sx), fy = floorf(sy), fz = floorf(sz);
        float tx = sx - fx, ty = sy - fy, tz = sz - fz;
        uint32_t bx = (uint32_t)fx, by = (uint32_t)fy, bz = (uint32_t)fz;
        h000 = ((bx * P1 + by) * P2 + bz) * P3;          // base-corner hash (pre-mask)
        float ux = 1.0f - tx, uy = 1.0f - ty, uz = 1.0f - tz;
        w[0] = ux * uy * uz;  w[1] = tx * uy * uz;
        w[2] = ux * ty * uz;  w[3] = tx * ty * uz;
        w[4] = ux * uy * tz;  w[5] = tx * uy * tz;
        w[6] = ux * ty * tz;  w[7] = tx * ty * tz;
    };

    // Levels 0..4 : distinct resolutions (16..256), all tables resident in LDS
#pragma unroll
    for (int lev = 0; lev < 5; ++lev) {
        uint32_t h000; float w[8];
        setup((float)(16 << lev), h000, w);
        const uint32_t mask = (2u << lev) - 1u;          // SIZES[lev]-1
        const uint32_t rb   = (2u << lev) - 2u;          // OFFS[lev]
        float o0 = 0.f, o1 = 0.f;
#pragma unroll
        for (int c = 0; c < 8; ++c) {
            uint32_t idx = rb + ((h000 + dC[c]) & mask);
            float2 f = smem[idx];                        // ds_load_b64
            o0 = fmaf(w[c], f.x, o0);
            o1 = fmaf(w[c], f.y, o1);
        }
        o[lev][0] = o0; o[lev][1] = o1;
    }

    // Levels 5..15 : all clamp to res=512 -> one hash/weight setup for 11 levels
    {
        uint32_t h000; float w[8];
        setup(512.0f, h000, w);
        const float2* __restrict__ tab2 = (const float2*)table;
#pragma unroll
        for (int lev = 5; lev < NLVL; ++lev) {
            const uint32_t mask = (2u << lev) - 1u;
            const uint32_t rb   = (2u << lev) - 2u;
            float o0 = 0.f, o1 = 0.f;
            if (lev < LDS_LEVELS) {
#pragma unroll
                for (int c = 0; c < 8; ++c) {
                    uint32_t idx = rb + ((h000 + dC[c]) & mask);
                    float2 f = smem[idx];                // LDS-resident levels
                    o0 = fmaf(w[c], f.x, o0);
                    o1 = fmaf(w[c], f.y, o1);
                }
            } else {
#pragma unroll
                for (int c = 0; c < 8; ++c) {
                    uint32_t idx = rb + ((h000 + dC[c]) & mask);
                    float2 f = tab2[idx];                // levels 14/15: L2-resident
                    o0 = fmaf(w[c], f.x, o0);
                    o1 = fmaf(w[c], f.y, o1);
                }
            }
            o[lev][0] = o0; o[lev][1] = o1;
        }
    }

    // 32 contiguous output floats per point; streaming 128-bit stores (write-once data)
    v4f_t* __restrict__ ov = (v4f_t*)(out + (size_t)p * 32u);
#pragma unroll
    for (int i = 0; i < 8; ++i) {
        v4f_t v;
        v.x = o[2 * i][0];     v.y = o[2 * i][1];
        v.z = o[2 * i + 1][0]; v.w = o[2 * i + 1][1];
        __builtin_nontemporal_store(v, ov + i);
    }
}

extern "C" void kernel_launch(void* const* d_in, const int* in_sizes, int n_in,
                              void* d_out, int out_size, void* d_ws, size_t ws_size,
                              hipStream_t stream) {
    (void)in_sizes; (void)n_in; (void)out_size; (void)d_ws; (void)ws_size;
    const float* coords = (const float*)d_in[0];   // (1048576, 3) f32
    const float* table  = (const float*)d_in[1];   // (131070, 2)  f32
    float* out = (float*)d_out;                    // (1048576, 32) f32
    dim3 grid(NPTS / BLOCK);
    hipLaunchKernelGGL(ngp_encode_kernel, grid, dim3(BLOCK), 0, stream,
                       coords, table, out);
}